// GCN_Net_38156489457767
// MI455X (gfx1250) — compile-verified
//
#include <hip/hip_runtime.h>
#include <math.h>

#define N_FEAT 128
#define N_HID  64
#define N_CLS  16

typedef float v2f __attribute__((ext_vector_type(2)));
typedef float v8f __attribute__((ext_vector_type(8)));

// ---------------- graph prep ----------------

__global__ void k_init_deg(float* __restrict__ deg, int n) {
    int i = blockIdx.x * blockDim.x + threadIdx.x;
    if (i < n) deg[i] = 1.0f;  // self-loop contributes 1 to every node's degree
}

__global__ void k_deg_atomic(const long long* __restrict__ dst, float* __restrict__ deg, int E) {
    int e = blockIdx.x * blockDim.x + threadIdx.x;
    if (e < E) atomicAdd(&deg[(int)dst[e]], 1.0f);
}

__global__ void k_dinv(float* __restrict__ deg, int n) {
    int i = blockIdx.x * blockDim.x + threadIdx.x;
    if (i < n) deg[i] = rsqrtf(deg[i]);  // deg >= 1 always (self loop)
}

__global__ void k_norm(const long long* __restrict__ src, const long long* __restrict__ dst,
                       const float* __restrict__ dinv, float* __restrict__ norm, int E) {
    int e = blockIdx.x * blockDim.x + threadIdx.x;
    if (e < E) norm[e] = dinv[(int)src[e]] * dinv[(int)dst[e]];
}

// ---------------- WMMA GEMMs (f32 16x16x4) ----------------
// One wave computes a 16-row tile. A-frag: lanes 0-15 carry K=k..k+1 of row
// M=lane, lanes 16-31 carry K=k+2..k+3. B-frag symmetric over K. C/D: 8 VGPRs,
// lane<16 -> M=v, lane>=16 -> M=8+v, N=lane%16.

__device__ __forceinline__ void wmma_store_tile(float* __restrict__ out, const v8f& acc,
                                                int m_base, int col, int ldm, int n_rows) {
#pragma unroll
    for (int v = 0; v < 8; ++v) {
        int m = m_base + v;
        if (m < n_rows) out[(long long)m * ldm + col] = acc[v];
    }
}

// H[n_rows x 64] = X[n_rows x 128] @ W[128 x 64]
__global__ __launch_bounds__(256)
void k_gemm1(const float* __restrict__ X, const float* __restrict__ W,
             float* __restrict__ H, int n_rows) {
    const int lane  = threadIdx.x & 31;
    const int wave  = threadIdx.x >> 5;
    const int mt    = blockIdx.x * 8 + wave;           // 16-row tile index
    if (mt * 16 >= n_rows) return;                      // wave-uniform: EXEC stays all-1
    const int lrow  = lane & 15;
    const int khalf = lane >> 4;                        // 0 or 1
    const int row   = min(mt * 16 + lrow, n_rows - 1);  // clamp loads, keep EXEC full

    v8f acc[4] = {v8f{}, v8f{}, v8f{}, v8f{}};
#pragma unroll 4
    for (int k = 0; k < N_FEAT; k += 4) {
        const int kk = k + khalf * 2;
        v2f a;
        a.x = X[(long long)row * N_FEAT + kk];
        a.y = X[(long long)row * N_FEAT + kk + 1];
#pragma unroll
        for (int nt = 0; nt < 4; ++nt) {
            const int col = nt * 16 + lrow;
            v2f b;
            b.x = W[kk * N_HID + col];
            b.y = W[(kk + 1) * N_HID + col];
            acc[nt] = __builtin_amdgcn_wmma_f32_16x16x4_f32(
                false, a, false, b, (short)0, acc[nt], false, false);
        }
    }
    const int m_base = mt * 16 + khalf * 8;
#pragma unroll
    for (int nt = 0; nt < 4; ++nt)
        wmma_store_tile(H, acc[nt], m_base, nt * 16 + lrow, N_HID, n_rows);
}

// H2[n_rows x 16] = relu(A[n_rows x 64]) @ W[64 x 16]   (ReLU fused into A load)
__global__ __launch_bounds__(256)
void k_gemm2(const float* __restrict__ A, const float* __restrict__ W,
             float* __restrict__ H, int n_rows) {
    const int lane  = threadIdx.x & 31;
    const int wave  = threadIdx.x >> 5;
    const int mt    = blockIdx.x * 8 + wave;
    if (mt * 16 >= n_rows) return;
    const int lrow  = lane & 15;
    const int khalf = lane >> 4;
    const int row   = min(mt * 16 + lrow, n_rows - 1);

    v8f acc = v8f{};
#pragma unroll 4
    for (int k = 0; k < N_HID; k += 4) {
        const int kk = k + khalf * 2;
        v2f a, b;
        a.x = fmaxf(A[(long long)row * N_HID + kk], 0.0f);
        a.y = fmaxf(A[(long long)row * N_HID + kk + 1], 0.0f);
        b.x = W[kk * N_CLS + lrow];
        b.y = W[(kk + 1) * N_CLS + lrow];
        acc = __builtin_amdgcn_wmma_f32_16x16x4_f32(
            false, a, false, b, (short)0, acc, false, false);
    }
    wmma_store_tile(H, acc, mt * 16 + khalf * 8, lrow, N_CLS, n_rows);
}

// ---------------- scatter-add message passing ----------------

// out[n][f] = bias[f] + dinv[n]^2 * H[n][f]  (self-loop term + bias, also inits out)
__global__ void k_self_bias(const float* __restrict__ H, const float* __restrict__ dinv,
                            const float* __restrict__ bias, float* __restrict__ out,
                            int n, int F) {
    long long i = (long long)blockIdx.x * blockDim.x + threadIdx.x;
    if (i < (long long)n * F) {
        int node = (int)(i / F), f = (int)(i % F);
        float di = dinv[node];
        out[i] = bias[f] + di * di * H[i];
    }
}

// Layer 1: F=64, one thread handles 4 consecutive floats of one edge message.
__global__ void k_edge64(const long long* __restrict__ src, const long long* __restrict__ dst,
                         const float* __restrict__ norm, const float* __restrict__ H,
                         float* __restrict__ out, int E) {
    long long t = (long long)blockIdx.x * blockDim.x + threadIdx.x;
    int e = (int)(t >> 4);
    int c = (int)(t & 15) * 4;
    if (e < E) {
        int s = (int)src[e], d = (int)dst[e];
        float w = norm[e];
        const float4 h = *(const float4*)&H[(long long)s * N_HID + c];
        float* o = &out[(long long)d * N_HID + c];
        atomicAdd(o + 0, w * h.x);
        atomicAdd(o + 1, w * h.y);
        atomicAdd(o + 2, w * h.z);
        atomicAdd(o + 3, w * h.w);
    }
}

// Layer 2: F=16, one thread handles 4 consecutive floats of one edge message.
__global__ void k_edge16(const long long* __restrict__ src, const long long* __restrict__ dst,
                         const float* __restrict__ norm, const float* __restrict__ H,
                         float* __restrict__ out, int E) {
    long long t = (long long)blockIdx.x * blockDim.x + threadIdx.x;
    int e = (int)(t >> 2);
    int c = (int)(t & 3) * 4;
    if (e < E) {
        int s = (int)src[e], d = (int)dst[e];
        float w = norm[e];
        const float4 h = *(const float4*)&H[(long long)s * N_CLS + c];
        float* o = &out[(long long)d * N_CLS + c];
        atomicAdd(o + 0, w * h.x);
        atomicAdd(o + 1, w * h.y);
        atomicAdd(o + 2, w * h.z);
        atomicAdd(o + 3, w * h.w);
    }
}

// In-place log-softmax over 16 classes per node.
__global__ void k_logsoftmax(float* __restrict__ io, int n) {
    int i = blockIdx.x * blockDim.x + threadIdx.x;
    if (i < n) {
        float r[N_CLS];
        float* p = &io[(long long)i * N_CLS];
        float mx = -3.402823466e+38f;
#pragma unroll
        for (int c = 0; c < N_CLS; ++c) { r[c] = p[c]; mx = fmaxf(mx, r[c]); }
        float s = 0.0f;
#pragma unroll
        for (int c = 0; c < N_CLS; ++c) s += __expf(r[c] - mx);
        float l = __logf(s);
#pragma unroll
        for (int c = 0; c < N_CLS; ++c) p[c] = r[c] - mx - l;
    }
}

// ---------------- launch ----------------

extern "C" void kernel_launch(void* const* d_in, const int* in_sizes, int n_in,
                              void* d_out, int out_size, void* d_ws, size_t ws_size,
                              hipStream_t stream) {
    const float*     x    = (const float*)d_in[0];
    const long long* ei   = (const long long*)d_in[1];   // int64 per reference
    const float*     W1   = (const float*)d_in[2];
    const float*     b1   = (const float*)d_in[3];
    const float*     W2   = (const float*)d_in[4];
    const float*     b2   = (const float*)d_in[5];
    float*           out  = (float*)d_out;

    const int N = in_sizes[0] / N_FEAT;    // 100000
    const int E = in_sizes[1] / 2;         // 1600000
    const long long* src = ei;
    const long long* dst = ei + E;

    // workspace carve (all offsets 16B aligned)
    float* ws   = (float*)d_ws;
    float* deg  = ws;                               // N   (becomes dinv in place)
    float* nrm  = deg + N;                          // E
    float* H1   = nrm + E;                          // N*64
    float* out1 = H1 + (long long)N * N_HID;        // N*64
    float* H2   = out1 + (long long)N * N_HID;      // N*16

    const int T = 256;
    // degree + normalization
    k_init_deg  <<<(N + T - 1) / T, T, 0, stream>>>(deg, N);
    k_deg_atomic<<<(E + T - 1) / T, T, 0, stream>>>(dst, deg, E);
    k_dinv      <<<(N + T - 1) / T, T, 0, stream>>>(deg, N);
    k_norm      <<<(E + T - 1) / T, T, 0, stream>>>(src, dst, deg, nrm, E);

    // layer 1
    const int mt1 = (N + 15) / 16;
    k_gemm1<<<(mt1 + 7) / 8, 256, 0, stream>>>(x, W1, H1, N);
    {
        long long tot = (long long)N * N_HID;
        k_self_bias<<<(int)((tot + T - 1) / T), T, 0, stream>>>(H1, deg, b1, out1, N, N_HID);
        long long et = (long long)E * 16;
        k_edge64<<<(int)((et + T - 1) / T), T, 0, stream>>>(src, dst, nrm, H1, out1, E);
    }

    // layer 2 (ReLU fused into GEMM2 A-load)
    k_gemm2<<<(mt1 + 7) / 8, 256, 0, stream>>>(out1, W2, H2, N);
    {
        long long tot = (long long)N * N_CLS;
        k_self_bias<<<(int)((tot + T - 1) / T), T, 0, stream>>>(H2, deg, b2, out, N, N_CLS);
        long long et = (long long)E * 4;
        k_edge16<<<(int)((et + T - 1) / T), T, 0, stream>>>(src, dst, nrm, H2, out, E);
    }

    // log-softmax in place on d_out
    k_logsoftmax<<<(N + T - 1) / T, T, 0, stream>>>(out, N);
}